// PhaseLoss_13022340841792
// MI455X (gfx1250) — compile-verified
//
#include <hip/hip_runtime.h>
#include <hip/hip_bf16.h>

// Problem geometry
#define NBATCH 32
#define NFREQ  513
#define NTIME  512
#define NTOTAL (NBATCH * NFREQ * NTIME)   // 8,404,992

// Tiling
#define TF 32                 // freq rows per tile
#define TT 64                 // time cols per tile (multiple of 4 -> b128 loads)
#define TILES_F 17            // ceil(513/32)
#define TILES_T 8             // 512/64
#define NTILES (NBATCH * TILES_F * TILES_T)  // 4352
#define NBLK 2176             // each block handles exactly 2 tiles
#define ROWS (TF + 1)         // 33 rows (one freq-halo row)
#define RSTRIDE 68            // floats per LDS row: 64 interior + halo@[64], mult of 4

typedef float v2f __attribute__((ext_vector_type(2)));
typedef float v8f __attribute__((ext_vector_type(8)));

// |x - 2pi*round(x/2pi)|  (rintf == round-half-even, matches jnp.round)
__device__ __forceinline__ float unwrap_abs(float x) {
    float k = rintf(x * 0.15915494309189535f);      // 1/(2pi)
    return fabsf(fmaf(k, -6.283185307179586f, x));  // x - 2pi*k
}

// (phase_ref - phase_est) mod 2pi via angle(ref * conj(est)).
// All three loss terms are invariant under per-element +2pi*k shifts,
// so this single atan2 is exact where the reference uses two.
__device__ __forceinline__ float phase_diff(float gr, float gi, float rr, float ri) {
    float re = fmaf(rr, gr, ri * gi);
    float im = fmaf(ri, gr, -(rr * gi));
    return atan2f(im, re);
}

// Exact f32 sum of all 32 lanes using V_WMMA_F32_16X16X4_F32.
// A(16x4): VGPR0 = lane sum, VGPR1 = 0  -> each lane sum appears exactly once.
// B(4x16): all ones. D[m][n] = s_m + s_{m+16}; fold 8 D VGPRs + cross-half shfl.
__device__ __forceinline__ float wave_reduce_wmma(float s) {
    v2f a; a.x = s;    a.y = 0.0f;
    v2f b; b.x = 1.0f; b.y = 1.0f;
    v8f c = {0.0f, 0.0f, 0.0f, 0.0f, 0.0f, 0.0f, 0.0f, 0.0f};
    c = __builtin_amdgcn_wmma_f32_16x16x4_f32(
            /*neg_a=*/false, a, /*neg_b=*/false, b,
            /*c_mod=*/(short)0, c, /*reuse_a=*/false, /*reuse_b=*/false);
    float u = ((c[0] + c[1]) + (c[2] + c[3])) + ((c[4] + c[5]) + (c[6] + c[7]));
    u += __shfl_xor(u, 16, 32);   // lanes 0-15 hold rows 0-7, 16-31 hold rows 8-15
    return u;
}

__global__ __launch_bounds__(256)
void phase_loss_partial(const float* __restrict__ gr, const float* __restrict__ gi,
                        const float* __restrict__ rr, const float* __restrict__ ri,
                        float* __restrict__ partial) {
    __shared__ float dbuf[ROWS * RSTRIDE];  // ~8.8 KB of phase-diff tile (+halos)
    __shared__ float wred[8];
    const int tid = threadIdx.x;
    float acc = 0.0f;

    for (int tile = blockIdx.x; tile < NTILES; tile += gridDim.x) {
        const int b  = tile / (TILES_F * TILES_T);
        const int r  = tile - b * (TILES_F * TILES_T);
        const int f0 = (r / TILES_T) * TF;
        const int t0 = (r % TILES_T) * TT;

        // ---- Fill interior: rows lf=0..32 map to f = f0-1 .. f0+31,
        //      cols 0..63 map to t = t0 .. t0+63. float4 (b128) loads. ----
        for (int idx = tid; idx < ROWS * (TT / 4); idx += 256) {
            const int lf = idx >> 4;      // TT/4 == 16
            const int q  = idx & 15;
            const int f  = f0 + lf - 1;
            float4 dv = {0.0f, 0.0f, 0.0f, 0.0f};   // virtual d = 0 at f < 0
            if (f >= 0 && f < NFREQ) {
                const int off = (b * NFREQ + f) * NTIME + t0 + q * 4;
                float4 a0 = *(const float4*)(gr + off);
                float4 a1 = *(const float4*)(gi + off);
                float4 a2 = *(const float4*)(rr + off);
                float4 a3 = *(const float4*)(ri + off);
                dv.x = phase_diff(a0.x, a1.x, a2.x, a3.x);
                dv.y = phase_diff(a0.y, a1.y, a2.y, a3.y);
                dv.z = phase_diff(a0.z, a1.z, a2.z, a3.z);
                dv.w = phase_diff(a0.w, a1.w, a2.w, a3.w);
            }
            *(float4*)&dbuf[lf * RSTRIDE + q * 4] = dv;   // 16B-aligned LDS store
        }
        // ---- Time-halo column (t = t0-1) stored at row offset TT ----
        for (int lf = tid; lf < ROWS; lf += 256) {
            const int f = f0 + lf - 1;
            float d = 0.0f;                          // virtual d = 0 at t < 0
            if (f >= 0 && f < NFREQ && t0 > 0) {
                const int off = (b * NFREQ + f) * NTIME + (t0 - 1);
                d = phase_diff(gr[off], gi[off], rr[off], ri[off]);
            }
            dbuf[lf * RSTRIDE + TT] = d;
        }
        __syncthreads();

        // ---- Accumulate all three unwrap terms for the 32x64 output tile.
        //      Zero halos make the f=0 / t=0 "-x[...,0]" terms come out
        //      automatically since unwrap(-x) == unwrap(x). ----
        for (int k = 0; k < 8; ++k) {
            const int idx = tid + (k << 8);       // 0..2047
            const int lf  = (idx >> 6) + 1;       // 1..32
            const int j   = idx & 63;             // 0..63
            const int f   = f0 + lf - 1;
            if (f < NFREQ) {
                const float dc = dbuf[lf * RSTRIDE + j];
                const float du = dbuf[(lf - 1) * RSTRIDE + j];
                const int   jl = (j == 0) ? TT : (j - 1);
                const float dl = dbuf[lf * RSTRIDE + jl];
                acc += unwrap_abs(dc)            // IP term
                     + unwrap_abs(du - dc)       // GD term (freq diff)
                     + unwrap_abs(dl - dc);      // PTD term (time diff)
            }
        }
        __syncthreads();
    }

    // ---- Block reduction: WMMA within each wave32, LDS across 8 waves ----
    const float wsum = wave_reduce_wmma(acc);
    if ((tid & 31) == 0) wred[tid >> 5] = wsum;
    __syncthreads();
    if (tid == 0) {
        float s = 0.0f;
        for (int w = 0; w < 8; ++w) s += wred[w];
        partial[blockIdx.x] = s;
    }
}

__global__ __launch_bounds__(256)
void phase_loss_final(const float* __restrict__ partial, float* __restrict__ out) {
    __shared__ float wred[8];
    const int tid = threadIdx.x;
    float s = 0.0f;
    for (int i = tid; i < NBLK; i += 256) s += partial[i];   // fixed order: deterministic
    const float wsum = wave_reduce_wmma(s);
    if ((tid & 31) == 0) wred[tid >> 5] = wsum;
    __syncthreads();
    if (tid == 0) {
        float t = 0.0f;
        for (int w = 0; w < 8; ++w) t += wred[w];
        out[0] = t * (1.0f / (float)NTOTAL);   // all three means share N
    }
}

extern "C" void kernel_launch(void* const* d_in, const int* in_sizes, int n_in,
                              void* d_out, int out_size, void* d_ws, size_t ws_size,
                              hipStream_t stream) {
    const float* gr = (const float*)d_in[0];  // spec_est_real
    const float* gi = (const float*)d_in[1];  // spec_est_imag
    const float* rr = (const float*)d_in[2];  // spec_ref_real
    const float* ri = (const float*)d_in[3];  // spec_ref_imag
    float* partials = (float*)d_ws;           // NBLK floats = 8.7 KB scratch

    phase_loss_partial<<<NBLK, 256, 0, stream>>>(gr, gi, rr, ri, partials);
    phase_loss_final<<<1, 256, 0, stream>>>(partials, (float*)d_out);
}